// Static_Plus_Influence_Model_86449101734282
// MI455X (gfx1250) — compile-verified
//
#include <hip/hip_runtime.h>

#define YY 5
#define NN 200000
#define GG 128
#define II 128
#define RR 3
#define BB 1024
#define KK 64
#define KDIM (RR * GG)     /* 384 contraction dim */
#define MROWS (YY * BB)    /* 5120 output rows   */

typedef __attribute__((ext_vector_type(2))) float v2f;
typedef __attribute__((ext_vector_type(8))) float v8f;

// ---------------------------------------------------------------------------
// Kernel 1: Weff[r,g,i] = (r==0) ? sum_j W_cite[j,g,i] : W_rel[r,g,i]
// Flat [384,128] row-major: row = r*128+g, col = i.
// ---------------------------------------------------------------------------
__global__ void build_weff(const float* __restrict__ W_rel,
                           const float* __restrict__ W_cite,
                           float* __restrict__ Weff) {
  int i = blockIdx.x * blockDim.x + threadIdx.x;
  if (i >= RR * GG * II) return;
  float v;
  if (i < GG * II) {
    v = W_cite[i] + W_cite[GG * II + i] + W_cite[2 * GG * II + i];
  } else {
    v = W_rel[i];
  }
  Weff[i] = v;
}

// ---------------------------------------------------------------------------
// Kernel 2: masked gather-mean.  One block per (y,r,b); thread = channel g.
// Each neighbor row = one coalesced 512B read.  Neighbor indices are
// block-uniform -> scalar loads.  4x unroll keeps loads in flight.
// Writes mean in [y*B+b][r*128+g] layout (contiguous contraction dim).
// ---------------------------------------------------------------------------
__global__ void gather_mean(const float* __restrict__ emb,
                            const long long* __restrict__ neigh,
                            const long long* __restrict__ counts,
                            float* __restrict__ mean) {
  int blk = blockIdx.x;            // (y*RR + r)*BB + b
  int b   = blk % BB;
  int t   = blk / BB;
  int r   = t % RR;
  int y   = t / RR;
  int g   = threadIdx.x;           // 0..127

  int c = (int)counts[blk];
  const long long* np = neigh + (long long)blk * KK;
  const float* eb = emb + (long long)y * NN * GG + g;

  float acc = 0.f;
  int k = 0;
  for (; k + 4 <= c; k += 4) {
    long long n0 = np[k], n1 = np[k + 1], n2 = np[k + 2], n3 = np[k + 3];
    float e0 = eb[n0 * GG];
    float e1 = eb[n1 * GG];
    float e2 = eb[n2 * GG];
    float e3 = eb[n3 * GG];
    acc += e0; acc += e1; acc += e2; acc += e3;
  }
  for (; k < c; ++k) acc += eb[np[k] * GG];

  float denom = (float)(c > 0 ? c : 1);
  mean[(long long)(y * BB + b) * KDIM + r * GG + g] = acc / denom;
}

// ---------------------------------------------------------------------------
// Kernel 3: C[5120,128] = A[5120,384] @ W[384,128] via V_WMMA_F32_16X16X4_F32.
// One wave per 16x16 tile; tiles = 320 x 8 = 2560 waves = 640 blocks x 128 thr.
// A 16x4 layout: lane L(0..15): v0=K0,v1=K1 ; lane L+16: v0=K2,v1=K3 (M=L&15).
// B 4x16 layout: lanes0-15: v0=row K0,v1=K1 ; lanes16-31: v0=K2,v1=K3 (N=L&15).
// C/D: vgpr v: lanes0-15 -> M=v, lanes16-31 -> M=v+8, N=L&15.
// ---------------------------------------------------------------------------
__global__ void wmma_gemm(const float* __restrict__ A,
                          const float* __restrict__ W,
                          float* __restrict__ C) {
  int lane  = threadIdx.x & 31;
  int wave  = blockIdx.x * (blockDim.x >> 5) + (threadIdx.x >> 5);
  int tm    = wave >> 3;                 // 0..319
  int tn    = wave & 7;                  // 0..7
  int mi    = lane & 15;
  int khalf = (lane >> 4) << 1;          // 0 or 2

  const float* Arow = A + (long long)(tm * 16 + mi) * KDIM + khalf;
  const float* Bcol = W + (long long)khalf * II + (tn * 16 + mi);

  v8f acc = {0.f, 0.f, 0.f, 0.f, 0.f, 0.f, 0.f, 0.f};

#pragma unroll 4
  for (int kk = 0; kk < KDIM; kk += 4) {
    v2f a, bv;
    a.x  = Arow[kk];
    a.y  = Arow[kk + 1];
    bv.x = Bcol[kk * II];
    bv.y = Bcol[(kk + 1) * II];
    // D = A(16x4) * B(4x16) + C : 8 args (neg_a, A, neg_b, B, c_mod, C, reuse_a, reuse_b)
    acc = __builtin_amdgcn_wmma_f32_16x16x4_f32(false, a, false, bv,
                                                (short)0, acc, false, false);
  }

  int ncol  = tn * 16 + mi;
  int mbase = tm * 16 + ((lane >> 4) << 3);
#pragma unroll
  for (int v = 0; v < 8; ++v) {
    C[(long long)(mbase + v) * II + ncol] = acc[v];
  }
}

// ---------------------------------------------------------------------------
extern "C" void kernel_launch(void* const* d_in, const int* in_sizes, int n_in,
                              void* d_out, int out_size, void* d_ws, size_t ws_size,
                              hipStream_t stream) {
  (void)in_sizes; (void)n_in; (void)out_size; (void)ws_size;

  const float*     emb    = (const float*)d_in[0];
  const float*     W_rel  = (const float*)d_in[1];
  const float*     W_cite = (const float*)d_in[2];
  const long long* neigh  = (const long long*)d_in[3];
  const long long* counts = (const long long*)d_in[4];
  float*           out    = (float*)d_out;

  // Workspace carve-up: Weff (384*128 f32 = 192KB, 256B-aligned), then mean
  // (5120*384 f32 = 7.5MB). Both fully rewritten every call.
  size_t weff_bytes = ((size_t)RR * GG * II * sizeof(float) + 255) & ~(size_t)255;
  float* Weff = (float*)d_ws;
  float* mean = (float*)((char*)d_ws + weff_bytes);

  build_weff<<<(RR * GG * II + 255) / 256, 256, 0, stream>>>(W_rel, W_cite, Weff);
  gather_mean<<<YY * RR * BB, GG, 0, stream>>>(emb, neigh, counts, mean);

  int tiles  = (MROWS / 16) * (II / 16);     // 2560 waves
  int blocks = tiles / 4;                    // 4 waves (128 thr) per block
  wmma_gemm<<<blocks, 128, 0, stream>>>(mean, Weff, out);
}